// CNF3_52931176956416
// MI455X (gfx1250) — compile-verified
//
#include <hip/hip_runtime.h>
#include <hip/hip_bf16.h>

#define D_IN   128      // in/out dim
#define H_DIM  256      // hidden dim
#define B_N    8192     // batch
#define BN_EPS 1e-5f

typedef __attribute__((ext_vector_type(2))) float v2f;
typedef __attribute__((ext_vector_type(8))) float v8f;

// ---------------------------------------------------------------------------
// GEMM (NT): out[m,n] = sum_k A[m*lda+k] * W[n*ldw+k] + bias[n]
// fp32 WMMA: v_wmma_f32_16x16x4_f32, wave32.
// Block = 256 threads = 8 waves arranged 4(M) x 2(N); block tile 64x32.
// A frag (16x4 f32): lanes 0-15 -> M=lane, K={kk,kk+1}; lanes 16-31 -> K={kk+2,kk+3}
// B frag (4x16 f32): lanes 0-15 -> N=lane, K={kk,kk+1}; lanes 16-31 -> K={kk+2,kk+3}
// D frag: VGPR v -> M = v + (lane>=16 ? 8 : 0), N = lane&15
// ---------------------------------------------------------------------------
template<int KDIM>
__global__ __launch_bounds__(256)
void wmma_gemm_nt(const float* __restrict__ A, const float* __restrict__ W,
                  const float* __restrict__ bias, float* __restrict__ out,
                  int lda, int ldw, int ldo)
{
    const int lane  = threadIdx.x & 31;
    const int wave  = threadIdx.x >> 5;
    const int wm    = wave & 3;            // wave position in M
    const int wn    = wave >> 2;           // wave position in N
    const int m0    = blockIdx.x * 64 + wm * 16;
    const int n0    = blockIdx.y * 32 + wn * 16;
    const int l15   = lane & 15;
    const int khalf = (lane >> 4) << 1;    // 0 for lanes 0-15, 2 for lanes 16-31

    const float* Ap = A + (size_t)(m0 + l15) * lda + khalf;
    const float* Wp = W + (size_t)(n0 + l15) * ldw + khalf;

    v8f acc = {};
#pragma unroll 8
    for (int kk = 0; kk < KDIM; kk += 4) {
        v2f af = *(const v2f*)(Ap + kk);
        v2f bf = *(const v2f*)(Wp + kk);
        // (neg_a, A, neg_b, B, c_mod, C, reuse_a, reuse_b)
        acc = __builtin_amdgcn_wmma_f32_16x16x4_f32(false, af, false, bf,
                                                    (short)0, acc, false, false);
    }

    const int   col = n0 + l15;
    const float bc  = bias[col];
    const int   mb  = m0 + ((lane >> 4) << 3);
#pragma unroll
    for (int v = 0; v < 8; ++v)
        out[(size_t)(mb + v) * ldo + col] = acc[v] + bc;
}

// ---------------------------------------------------------------------------
// Per-column batch stats over h[B,H]: mu[k], s[k] = rsqrt(var+eps). Deterministic.
// ---------------------------------------------------------------------------
__global__ __launch_bounds__(256)
void colstats_kernel(const float* __restrict__ h, float* __restrict__ mu,
                     float* __restrict__ sarr)
{
    const int k = blockIdx.x, t = threadIdx.x;
    float s = 0.f, q = 0.f;
    for (int i = t; i < B_N; i += 256) {
        float v = h[(size_t)i * H_DIM + k];
        s += v; q += v * v;
    }
    __shared__ float ls[256], lq[256];
    ls[t] = s; lq[t] = q;
    __syncthreads();
    for (int o = 128; o; o >>= 1) {
        if (t < o) { ls[t] += ls[t + o]; lq[t] += lq[t + o]; }
        __syncthreads();
    }
    if (t == 0) {
        float m   = ls[0] * (1.f / B_N);
        float var = lq[0] * (1.f / B_N) - m * m;
        mu[k]   = m;
        sarr[k] = rsqrtf(var + BN_EPS);
    }
}

// ---------------------------------------------------------------------------
// a = relu(gamma * (h - mu) * s + beta), vectorized float4 (4 cols per thread)
// ---------------------------------------------------------------------------
__global__ __launch_bounds__(256)
void bn_relu_kernel(const float* __restrict__ h, const float* __restrict__ mu,
                    const float* __restrict__ sarr, const float* __restrict__ gamma,
                    const float* __restrict__ beta, float* __restrict__ a)
{
    size_t idx = ((size_t)blockIdx.x * 256 + threadIdx.x) * 4;
    int    k   = (int)(idx & (H_DIM - 1));
    float4 hv  = *(const float4*)(h + idx);
    float4 mv  = *(const float4*)(mu + k);
    float4 sv  = *(const float4*)(sarr + k);
    float4 gv  = *(const float4*)(gamma + k);
    float4 bv  = *(const float4*)(beta + k);
    float4 av;
    av.x = fmaxf(gv.x * (hv.x - mv.x) * sv.x + bv.x, 0.f);
    av.y = fmaxf(gv.y * (hv.y - mv.y) * sv.y + bv.y, 0.f);
    av.z = fmaxf(gv.z * (hv.z - mv.z) * sv.z + bv.z, 0.f);
    av.w = fmaxf(gv.w * (hv.w - mv.w) * sv.w + bv.w, 0.f);
    *(float4*)(a + idx) = av;
}

// ---------------------------------------------------------------------------
// Per-column R[k] = sum_b 1{relu active}, Q[k] = sum_b r * hn. Deterministic.
// ---------------------------------------------------------------------------
__global__ __launch_bounds__(256)
void rq_kernel(const float* __restrict__ h, const float* __restrict__ a,
               const float* __restrict__ mu, const float* __restrict__ sarr,
               float* __restrict__ R, float* __restrict__ Q)
{
    const int k = blockIdx.x, t = threadIdx.x;
    const float m = mu[k], s = sarr[k];
    float r = 0.f, q = 0.f;
    for (int i = t; i < B_N; i += 256) {
        size_t off = (size_t)i * H_DIM + k;
        if (a[off] > 0.f) {
            r += 1.f;
            q += (h[off] - m) * s;
        }
    }
    __shared__ float lr[256], lq[256];
    lr[t] = r; lq[t] = q;
    __syncthreads();
    for (int o = 128; o; o >>= 1) {
        if (t < o) { lr[t] += lr[t + o]; lq[t] += lq[t + o]; }
        __syncthreads();
    }
    if (t == 0) { R[k] = lr[0]; Q[k] = lq[0]; }
}

// ---------------------------------------------------------------------------
// g[k] = diag(W1@W2)[k] * gamma[k] * s[k];  c1 = g;  cQ = g*Q/B;  C0 = sum g*R/B
// ---------------------------------------------------------------------------
__global__ __launch_bounds__(256)
void coef_kernel(const float* __restrict__ W1, const float* __restrict__ W2,
                 const float* __restrict__ gamma, const float* __restrict__ sarr,
                 const float* __restrict__ R, const float* __restrict__ Q,
                 float* __restrict__ c1, float* __restrict__ cQ, float* __restrict__ C0)
{
    const int k = threadIdx.x;
    float m = 0.f;
#pragma unroll 4
    for (int i = 0; i < D_IN; ++i)
        m += W1[k * D_IN + i] * W2[i * H_DIM + k];
    const float g    = m * gamma[k] * sarr[k];
    const float invB = 1.f / B_N;
    c1[k] = g;
    cQ[k] = g * Q[k] * invB;
    __shared__ float lc[256];
    lc[k] = g * R[k] * invB;
    __syncthreads();
    for (int o = 128; o; o >>= 1) {
        if (k < o) lc[k] += lc[k + o];
        __syncthreads();
    }
    if (k == 0) C0[0] = lc[0];
}

// ---------------------------------------------------------------------------
// dlogp[b] = -trace[b] = C0 - sum_k ( c1[k]*r[b,k] - cQ[k]*hn[b,k] )
// One wave32 per batch row; lane covers 8 consecutive k; shfl_xor reduction.
// ---------------------------------------------------------------------------
__global__ __launch_bounds__(256)
void trace_kernel(const float* __restrict__ h, const float* __restrict__ a,
                  const float* __restrict__ mu, const float* __restrict__ sarr,
                  const float* __restrict__ c1, const float* __restrict__ cQ,
                  const float* __restrict__ C0, float* __restrict__ dlogp)
{
    const int lane = threadIdx.x & 31;
    const int wave = threadIdx.x >> 5;
    const int b    = blockIdx.x * 8 + wave;
    const float* hr = h + (size_t)b * H_DIM;
    const float* ar = a + (size_t)b * H_DIM;
    float t = 0.f;
#pragma unroll
    for (int c = 0; c < 8; c += 4) {
        int k = lane * 8 + c;
        float4 hv  = *(const float4*)(hr + k);
        float4 av  = *(const float4*)(ar + k);
        float4 mv  = *(const float4*)(mu + k);
        float4 sv  = *(const float4*)(sarr + k);
        float4 c1v = *(const float4*)(c1 + k);
        float4 cqv = *(const float4*)(cQ + k);
        t += (av.x > 0.f ? c1v.x : 0.f) - cqv.x * (hv.x - mv.x) * sv.x;
        t += (av.y > 0.f ? c1v.y : 0.f) - cqv.y * (hv.y - mv.y) * sv.y;
        t += (av.z > 0.f ? c1v.z : 0.f) - cqv.z * (hv.z - mv.z) * sv.z;
        t += (av.w > 0.f ? c1v.w : 0.f) - cqv.w * (hv.w - mv.w) * sv.w;
    }
    for (int o = 16; o; o >>= 1) t += __shfl_xor(t, o, 32);
    if (lane == 0) dlogp[b] = C0[0] - t;
}

// ---------------------------------------------------------------------------
extern "C" void kernel_launch(void* const* d_in, const int* in_sizes, int n_in,
                              void* d_out, int out_size, void* d_ws, size_t ws_size,
                              hipStream_t stream)
{
    (void)in_sizes; (void)n_in; (void)out_size; (void)ws_size;
    // setup_inputs order: t, z, logp_z, W1, b1, gamma, beta, W2, b2
    const float* z     = (const float*)d_in[1];
    const float* W1    = (const float*)d_in[3];
    const float* b1    = (const float*)d_in[4];
    const float* gamma = (const float*)d_in[5];
    const float* beta  = (const float*)d_in[6];
    const float* W2    = (const float*)d_in[7];
    const float* b2    = (const float*)d_in[8];
    float*       out   = (float*)d_out;           // [B*D] dz_dt, then [B] dlogp

    // workspace layout
    float* h    = (float*)d_ws;                    // B*H
    float* a    = h  + (size_t)B_N * H_DIM;        // B*H
    float* mu   = a  + (size_t)B_N * H_DIM;        // H
    float* s    = mu + H_DIM;                      // H
    float* R    = s  + H_DIM;                      // H
    float* Q    = R  + H_DIM;                      // H
    float* c1   = Q  + H_DIM;                      // H
    float* cQ   = c1 + H_DIM;                      // H
    float* C0   = cQ + H_DIM;                      // 1

    dim3 blk(256);

    // h = z @ W1^T + b1        (fp32 WMMA)
    wmma_gemm_nt<D_IN><<<dim3(B_N / 64, H_DIM / 32), blk, 0, stream>>>(
        z, W1, b1, h, D_IN, D_IN, H_DIM);
    // batch stats
    colstats_kernel<<<H_DIM, blk, 0, stream>>>(h, mu, s);
    // a = relu(BN(h))
    bn_relu_kernel<<<(B_N * H_DIM) / 1024, blk, 0, stream>>>(h, mu, s, gamma, beta, a);
    // R, Q per column
    rq_kernel<<<H_DIM, blk, 0, stream>>>(h, a, mu, s, R, Q);
    // trace coefficients
    coef_kernel<<<1, blk, 0, stream>>>(W1, W2, gamma, s, R, Q, c1, cQ, C0);
    // dz_dt = a @ W2^T + b2    (fp32 WMMA) -> first B*D of d_out
    wmma_gemm_nt<H_DIM><<<dim3(B_N / 64, D_IN / 32), blk, 0, stream>>>(
        a, W2, b2, out, H_DIM, H_DIM, D_IN);
    // dlogp_z_dt -> tail of d_out
    trace_kernel<<<B_N / 8, blk, 0, stream>>>(h, a, mu, s, c1, cQ, C0,
                                              out + (size_t)B_N * D_IN);
}